// TTEmbedding_70875550318685
// MI455X (gfx1250) — compile-verified
//
#include <hip/hip_runtime.h>

// TT-embedding for MI455X (gfx1250, wave32).
// Per token: two tiny fp32 GEMMs via V_WMMA_F32_16X16X4_F32.
// Cores staged (and transposed) into LDS so every WMMA operand K-pair is a
// single ds_load_b64; zero-padding done by address-select against a zeroed
// LDS pad (no EXEC masking in the inner loop). Coalesced float4 output.

typedef float v2f __attribute__((ext_vector_type(2)));
typedef float v4f __attribute__((ext_vector_type(4)));
typedef float v8f __attribute__((ext_vector_type(8)));

#define N_C0   2560     // core0: (1,40,8,8)   staged as [i1][j1][r1]
#define N_C1   18432    // core1: (8,36,8,8)   staged transposed as [i2][n=j2*8+r2][r1]
#define N_C2   2240     // core2: (8,35,8,1)   staged transposed as [i3][j3][r2]
#define N_ZP   16       // zero pad for address-selected padding reads
#define EMB    512
#define WAVES_PER_BLOCK 8
#define BLOCK  (WAVES_PER_BLOCK * 32)

__global__ void __launch_bounds__(BLOCK)
tt_embed_wmma(const int* __restrict__ X,
              const float* __restrict__ C0,
              const float* __restrict__ C1,
              const float* __restrict__ C2,
              float* __restrict__ OUT,
              int ntok)
{
    extern __shared__ float lds[];
    float* sA = lds;                    // 2560 : core0 [i1*64 + j1*8 + r1]
    float* sB = sA + N_C0;              // 18432: core1T [i2*512 + n*8 + r1]
    float* sC = sB + N_C1;              // 2240 : core2T [i3*64 + j3*8 + r2]
    float* sZ = sC + N_C2;              // 16   : zeros

    const int tid  = threadIdx.x;
    const int warp = tid >> 5;
    const int lane = tid & 31;

    float* sM = sZ + N_ZP + warp * 1024;   // per-wave scratch: M[j1][n] = [8][64]
    float* sO = sM + 512;                  // per-wave scratch: out[p][j3] = [64][8]

    // ---- stage TT cores into LDS (once per block, transposing core1/core2) ----
    for (int i = tid; i < N_C0; i += BLOCK) sA[i] = C0[i];
    for (int i = tid; i < N_C1; i += BLOCK) {
        // src  [r1][i2][j2][r2]: r1 = i/2304, rem = i%2304 -> i2 = rem/64, n = rem%64
        const int r1  = i >> 11;            // i / 2048? no: 2304 not pow2 -> do exact
        (void)r1;
        const int r1e  = i / 2304;
        const int rem  = i - r1e * 2304;
        const int i2   = rem >> 6;
        const int n    = rem & 63;
        sB[i2 * 512 + n * 8 + r1e] = C1[i];
    }
    for (int i = tid; i < N_C2; i += BLOCK) {
        // src [r2][i3][j3]: r2 = i/280, rem -> i3 = rem/8, j3 = rem%8
        const int r2  = i / 280;
        const int rem = i - r2 * 280;
        const int i3  = rem >> 3;
        const int j3  = rem & 7;
        sC[i3 * 64 + j3 * 8 + r2] = C2[i];
    }
    if (tid < N_ZP) sZ[tid] = 0.0f;
    __syncthreads();

    const int half = lane >> 4;   // 0: K in {0,1}, 1: K in {2,3} (16x4 fp32 A/B layout)
    const int lm   = lane & 15;

    const int wave    = blockIdx.x * WAVES_PER_BLOCK + warp;
    const int wstride = gridDim.x * WAVES_PER_BLOCK;

    for (int tok = wave; tok < ntok; tok += wstride) {
        const int v   = X[tok];                 // wave-uniform token id
        const int i1  = v / 1260;               // 1260 = 36*35
        const int rem = v - i1 * 1260;
        const int i2  = rem / 35;
        const int i3  = rem - i2 * 35;

        const float* A0  = sA + i1 * 64;        // [j1*8 + r1]
        const float* B0t = sB + i2 * 512;       // [n*8 + r1]
        const float* C0t = sC + i3 * 64;        // [j3*8 + r2]

        // address-selected zero padding: lanes with lm>=8 read the zero pad
        const float* Arow = (lm < 8) ? (A0  + lm * 8) : sZ;   // stage-A A operand row
        const float* Crow = (lm < 8) ? (C0t + lm * 8) : sZ;   // stage-B B operand row

        // ---- stage A: M[j1][n], n = j2*8+r2, K = r1 (8) ----
        #pragma unroll
        for (int t = 0; t < 4; ++t) {
            v8f acc = {};
            const int n = t * 16 + lm;
            #pragma unroll
            for (int kk = 0; kk < 2; ++kk) {
                const int k0 = kk * 4 + half * 2;            // K pair {k0, k0+1}
                v2f a = *(const v2f*)(Arow + k0);            // one ds_load_b64
                v2f b = *(const v2f*)(B0t + n * 8 + k0);     // one ds_load_b64
                acc = __builtin_amdgcn_wmma_f32_16x16x4_f32(
                        false, a, false, b, (short)0, acc, false, false);
            }
            if (half == 0) {                    // valid rows j1 = 0..7 live in lanes 0-15
                #pragma unroll
                for (int g = 0; g < 8; ++g)
                    sM[g * 64 + t * 16 + lm] = acc[g];
            }
        }
        asm volatile("s_wait_dscnt 0" ::: "memory");  // wave-local LDS fence

        // ---- stage B: out[p][j3], p = j1*8+j2, K = r2 (8) ----
        #pragma unroll
        for (int t = 0; t < 4; ++t) {
            v8f acc = {};
            const int p = t * 16 + lm;                       // row = j1*8 + j2
            const float* Mrow = sM + (p >> 3) * 64 + (p & 7) * 8;  // [r2]
            #pragma unroll
            for (int kk = 0; kk < 2; ++kk) {
                const int k0 = kk * 4 + half * 2;
                v2f a = *(const v2f*)(Mrow + k0);            // one ds_load_b64
                v2f b = *(const v2f*)(Crow + k0);            // one ds_load_b64
                acc = __builtin_amdgcn_wmma_f32_16x16x4_f32(
                        false, a, false, b, (short)0, acc, false, false);
            }
            if (lm < 8) {                        // valid cols j3 = 0..7
                #pragma unroll
                for (int g = 0; g < 8; ++g)
                    sO[(t * 16 + g + 8 * half) * 8 + lm] = acc[g];
            }
        }
        asm volatile("s_wait_dscnt 0" ::: "memory");

        // ---- coalesced output: 512 floats, float4 per lane, padding mask ----
        const float scale = (v == 0) ? 0.0f : 1.0f;   // PADDING_IDX = 0
        float* outp = OUT + (size_t)tok * EMB;
        #pragma unroll
        for (int it = 0; it < 4; ++it) {
            const int o = it * 128 + lane * 4;
            v4f val = *(const v4f*)(sO + o);
            val.x *= scale; val.y *= scale; val.z *= scale; val.w *= scale;
            *(v4f*)(outp + o) = val;
        }
    }
}

extern "C" void kernel_launch(void* const* d_in, const int* in_sizes, int n_in,
                              void* d_out, int out_size, void* d_ws, size_t ws_size,
                              hipStream_t stream) {
    (void)n_in; (void)d_ws; (void)ws_size; (void)out_size;
    const int*   X  = (const int*)  d_in[0];
    const float* C0 = (const float*)d_in[1];
    const float* C1 = (const float*)d_in[2];
    const float* C2 = (const float*)d_in[3];
    float* OUT = (float*)d_out;
    const int ntok = in_sizes[0];   // 32*512 = 16384

    const size_t lds_bytes =
        (size_t)(N_C0 + N_C1 + N_C2 + N_ZP + WAVES_PER_BLOCK * 1024) * sizeof(float);

    static int attr_set = 0;  // host-side config only; deterministic
    if (!attr_set) {
        hipFuncSetAttribute((const void*)tt_embed_wmma,
                            hipFuncAttributeMaxDynamicSharedMemorySize,
                            (int)lds_bytes);
        attr_set = 1;
    }

    const int blocks = 256;   // 2048 waves, grid-stride over 16384 tokens
    tt_embed_wmma<<<blocks, BLOCK, lds_bytes, stream>>>(X, C0, C1, C2, OUT, ntok);
}